// Euler_86947317941191
// MI455X (gfx1250) — compile-verified
//
#include <hip/hip_runtime.h>
#include <stdint.h>

// Problem constants (from the reference): B=32, M=512, T=2048, k=1/48000
#define B_  32
#define M_  512
#define T_  2048
#define K_  (1.0f / 48000.0f)

typedef __attribute__((ext_vector_type(2))) float v2f;
typedef __attribute__((ext_vector_type(4))) float v4f;

// gfx1250 hardware tanh (CDNA5 TRANS op).
__device__ __forceinline__ float hw_tanh(float x) {
#if __has_builtin(__builtin_amdgcn_tanhf)
    return __builtin_amdgcn_tanhf(x);
#elif __has_builtin(__builtin_amdgcn_tanh_f32)
    return __builtin_amdgcn_tanh_f32(x);
#else
    float r;
    asm("v_tanh_f32 %0, %1" : "=v"(r) : "v"(x));
    return r;
#endif
}

// CDNA5 async DMA: copy an 8KB row (2048 floats) global -> LDS using
// GLOBAL_LOAD_ASYNC_TO_LDS_B128 (ASYNCcnt-tracked; no VGPR round-trip).
// Per the ISA, IOFFSET is added to BOTH the global and the LDS address, so
// the four offsets tile the row; each wave's 32 lanes x 16B cover 512B per
// instruction, and the block's 4 waves (tid 0..127) cover each 2KB chunk.
__device__ __forceinline__ void async_row_to_lds(uint32_t lds_addr,
                                                 uint32_t voff,
                                                 const float* gsrc) {
    asm volatile(
        "global_load_async_to_lds_b128 %0, %1, %2 offset:0\n\t"
        "global_load_async_to_lds_b128 %0, %1, %2 offset:2048\n\t"
        "global_load_async_to_lds_b128 %0, %1, %2 offset:4096\n\t"
        "global_load_async_to_lds_b128 %0, %1, %2 offset:6144"
        :
        : "v"(lds_addr), "v"(voff), "s"(gsrc)
        : "memory");
}

__device__ __forceinline__ void wait_async_zero() {
#if __has_builtin(__builtin_amdgcn_s_wait_asynccnt)
    __builtin_amdgcn_s_wait_asynccnt(0);
#else
    asm volatile("s_wait_asynccnt 0x0" ::: "memory");
#endif
}

// One block handles half a batch (256 oscillators); 2 blocks per batch,
// 64 blocks total. Each thread owns 2 adjacent oscillators -> b64 stores.
__global__ __launch_bounds__(128) void euler_traj_kernel(
    const float* __restrict__ y0,        // [B, 2M]
    const float* __restrict__ omega_sq,  // [B, M]
    const float* __restrict__ mu_sq,     // [B, 1]
    const float* __restrict__ sigma,     // [B, M]
    const float* __restrict__ Phi_e,     // [B, M]
    const float* __restrict__ fe,        // [B, T]
    float* __restrict__ out)             // [T, B, 2M]
{
    __shared__ float fe_s[T_];           // 8 KB: this batch's forcing signal

    const int tid = threadIdx.x;
    const int b   = blockIdx.x >> 1;          // batch
    const int m0  = ((blockIdx.x & 1) << 8)   // which half of M
                  + (tid << 1);               // 2 oscillators per thread

    // ---- kick off async DMA of fe[b, :] into LDS (overlaps loads below).
    // Flat LDS pointers carry the wave-relative LDS byte offset in addr[31:0].
    {
        const uint32_t lds_base = (uint32_t)(uintptr_t)(&fe_s[0]);
        async_row_to_lds(lds_base + (uint32_t)(tid << 4),
                         (uint32_t)(tid << 4),
                         fe + (size_t)b * T_);
    }

    // ---- per-thread state & parameters (live in VGPRs for all 2048 steps)
    const size_t bm = (size_t)b * M_ + m0;
    v2f q   = *(const v2f*)(y0 + (size_t)b * (2 * M_) + m0);
    v2f p   = *(const v2f*)(y0 + (size_t)b * (2 * M_) + M_ + m0);
    v2f w   = *(const v2f*)(omega_sq + bm);     // omega^2
    v2f s2  = 2.0f * *(const v2f*)(sigma + bm); // 2*sigma
    v2f phi = *(const v2f*)(Phi_e + bm);
    const float mus = mu_sq[b];
    const float k   = K_;

    // Each wave drains its own async queue, then the block barrier makes
    // every wave's LDS writes visible to all.
    wait_async_zero();
    __syncthreads();

    // q rows live at out[n][b][m0..], p rows at +M_
    float* oq = out + (size_t)b * (2 * M_) + m0;
    const size_t step_stride = (size_t)B_ * (2 * M_);   // 32768 floats / step

    for (int n = 0; n < T_; n += 4) {
        // uniform-address LDS read -> hardware broadcast, one b128 per 4 steps
        v4f fe4 = *(const v4f*)&fe_s[n];
        #pragma unroll
        for (int u = 0; u < 4; ++u) {
            const float fen = fe4[u];
            v2f nl;
            nl.x = hw_tanh(q.x);
            nl.y = hw_tanh(q.y);
            v2f q1 = q + k * p;
            // same association as reference: p + k*(-2s*p - w*q + mu*nl + phi*fe)
            v2f inner = -s2 * p - w * q + mus * nl + phi * fen;
            v2f p1 = p + k * inner;

            // trajectory is write-once, 256MB > 192MB L2: non-temporal stores
            __builtin_nontemporal_store(q1, (v2f*)oq);
            __builtin_nontemporal_store(p1, (v2f*)(oq + M_));

            q = q1;
            p = p1;
            oq += step_stride;
        }
    }
}

extern "C" void kernel_launch(void* const* d_in, const int* in_sizes, int n_in,
                              void* d_out, int out_size, void* d_ws, size_t ws_size,
                              hipStream_t stream) {
    const float* y0       = (const float*)d_in[0];
    const float* omega_sq = (const float*)d_in[1];
    const float* mu_sq    = (const float*)d_in[2];
    const float* sigma    = (const float*)d_in[3];
    const float* Phi_e    = (const float*)d_in[4];
    const float* fe       = (const float*)d_in[5];
    float* out            = (float*)d_out;

    dim3 grid(64);    // 2 blocks per batch x 32 batches
    dim3 block(128);  // 4 wave32 per block; 2 oscillators per thread
    hipLaunchKernelGGL(euler_traj_kernel, grid, block, 0, stream,
                       y0, omega_sq, mu_sq, sigma, Phi_e, fe, out);
}